// BagOfWords_22763326668852
// MI455X (gfx1250) — compile-verified
//
#include <hip/hip_runtime.h>

// Bag-of-words per-row bincount for MI455X (gfx1250).
//   in : [16384, 512] int32 token ids in [0, 1101)
//   out: [16384, 1100] float32 counts of token ids 1..1100
//
// Memory-bound scatter: ~104 MB total traffic -> ~4.5 us floor @ 23.3 TB/s.
// Strategy: 1 workgroup per row. Async-copy the row's tokens to LDS
// (ASYNCcnt path) while zeroing the LDS histogram, LDS ds_add_u32 scatter,
// then 16B-aligned b128 streaming stores of the converted row.

#define N_BINS    1100   // output bins = token ids 1..1100
#define SEQ       512
#define THREADS   256    // 8 wave32 per workgroup
#define VEC_CHUNKS (N_BINS / 4)  // 275 float4 per row (exact)

typedef unsigned int uintx4 __attribute__((ext_vector_type(4)));
typedef float        floatx4 __attribute__((ext_vector_type(4)));

__global__ __launch_bounds__(THREADS)
void bow_hist_kernel(const int* __restrict__ tokens, float* __restrict__ out)
{
    __shared__ __align__(16) unsigned int counts[N_BINS + 4]; // 4416 B, padded
    __shared__ __align__(16) int          tok[SEQ];           // 2048 B staging

    const int row = blockIdx.x;
    const int t   = threadIdx.x;

    // ---- 1) Kick off async global->LDS copy of this row's 512 tokens. ----
    // GVS mode: saddr = uniform row base (SGPR pair), vaddr = per-thread
    // byte offset, vdst VGPR = LDS byte address. 8 B per thread, 2 KB total.
    {
        unsigned long long row_base =
            (unsigned long long)(tokens + (size_t)row * SEQ);
        unsigned int voff    = (unsigned int)(t * 8);
        unsigned int lds_dst = (unsigned int)(unsigned long long)(&tok[2 * t]);
        asm volatile("global_load_async_to_lds_b64 %0, %1, %2"
                     :
                     : "v"(lds_dst), "v"(voff), "s"(row_base)
                     : "memory");
    }

    // ---- 2) Zero the LDS histogram while the async copy is in flight. ----
    {
        uintx4 z = (uintx4)0;
        #pragma unroll
        for (int i = t; i < VEC_CHUNKS; i += THREADS) {
            *(uintx4*)&counts[4 * i] = z;   // ds_store_b128, 16B aligned
        }
        // one thread clears the pad words
        if (t == 0) {
            counts[N_BINS + 0] = 0u; counts[N_BINS + 1] = 0u;
            counts[N_BINS + 2] = 0u; counts[N_BINS + 3] = 0u;
        }
    }

    // ---- 3) Wait for async tokens, sync the zeroed histogram. ----
    asm volatile("s_wait_asynccnt 0" ::: "memory");
    __syncthreads();

    // ---- 4) Scatter: each thread consumes its own 2 staged tokens. ----
    // Remap token v -> bin v-1 (token 0 is dropped by the reference's [:,1:]),
    // so the LDS histogram is exactly the output row, 16B aligned at bin 0.
    {
        int v0 = tok[2 * t + 0];
        int v1 = tok[2 * t + 1];
        if (v0 > 0) atomicAdd(&counts[v0 - 1], 1u);   // ds_add_u32 (no rtn)
        if (v1 > 0) atomicAdd(&counts[v1 - 1], 1u);
    }
    __syncthreads();

    // ---- 5) Convert + stream the row out with aligned b128 stores. ----
    // 1100 floats = 275 float4; 4400 B row pitch keeps every row 16B aligned.
    {
        float* __restrict__ orow = out + (size_t)row * N_BINS;
        #pragma unroll
        for (int i = t; i < VEC_CHUNKS; i += THREADS) {
            uintx4 c = *(const uintx4*)&counts[4 * i];  // ds_load_b128
            floatx4 f;
            f.x = (float)c.x; f.y = (float)c.y;
            f.z = (float)c.z; f.w = (float)c.w;
            *(floatx4*)&orow[4 * i] = f;                // global_store_b128
        }
    }
}

extern "C" void kernel_launch(void* const* d_in, const int* in_sizes, int n_in,
                              void* d_out, int out_size, void* d_ws, size_t ws_size,
                              hipStream_t stream)
{
    const int* tokens = (const int*)d_in[0];
    float*     out    = (float*)d_out;

    const int n_rows = in_sizes[0] / SEQ;   // 16384

    bow_hist_kernel<<<dim3(n_rows), dim3(THREADS), 0, stream>>>(tokens, out);
}